// SingleDisCoLoss_61727269978445
// MI455X (gfx1250) — compile-verified
//
#include <hip/hip_runtime.h>
#include <math.h>

// SingleDisCoLoss for MI455X (gfx1250, wave32).
// total = BCE(p,y,w) + 0.1 * dcorr^2 on the labels==0 subset.
// Pass-1 row sums are accumulated with V_WMMA_F32_16X16X4_F32 (matrix pipe,
// co-executes with the VALU that generates |x_i - x_j| fragments).

#define NELEM 8192
#define TJ 256

typedef float v2f __attribute__((ext_vector_type(2)));
typedef float v8f __attribute__((ext_vector_type(8)));

// ws float layout
#define WS_NF    0
#define WS_BCE   1
#define WS_ATOT  2
#define WS_BTOT  3
#define WS_PART  8              // 32 blocks * 3 partials
#define WS_A     256            // a_row (raw, then normalized in-place)
#define WS_B     (256 + NELEM)  // b_row

// ---------------- K0: nf + BCE mean (1 block) ----------------
__global__ void k0_scalars(const float* __restrict__ inf, const int* __restrict__ lab,
                           const float* __restrict__ w, float* __restrict__ ws) {
    __shared__ float s_b[256];
    __shared__ float s_n[256];
    int t = threadIdx.x;
    float bces = 0.0f, nfs = 0.0f;
    for (int i = t; i < NELEM; i += 256) {
        float p  = inf[i];
        float y  = (float)lab[i];
        float lp = fmaxf(logf(p), -100.0f);        // torch BCE clamps at -100
        float l1 = fmaxf(log1pf(-p), -100.0f);
        bces += w[i] * -(y * lp + (1.0f - y) * l1);
        nfs  += (lab[i] == 0) ? 1.0f : 0.0f;
    }
    s_b[t] = bces; s_n[t] = nfs;
    __syncthreads();
    for (int o = 128; o > 0; o >>= 1) {
        if (t < o) { s_b[t] += s_b[t + o]; s_n[t] += s_n[t + o]; }
        __syncthreads();
    }
    if (t == 0) { ws[WS_BCE] = s_b[0] / (float)NELEM; ws[WS_NF] = s_n[0]; }
}

// ---------------- K1: pass-1 row sums via WMMA f32 16x16x4 ----------------
// Each wave owns 16 rows i. A[m][k] = |x_{i_m} - x_{j_k}| * wm_{j_k}, B = ones,
// D += A x B accumulates a_row over all j in the matrix pipe.
__global__ void k1_rowsums(const float* __restrict__ inf, const int* __restrict__ lab,
                           const float* __restrict__ tgt, const float* __restrict__ w,
                           float* __restrict__ ws) {
    __shared__ float4 xtw[TJ];  // {x_j, t_j, wm_j, 0}
    int t    = threadIdx.x;
    int lane = t & 31;
    int wid  = (blockIdx.x * blockDim.x + t) >> 5;   // global wave id
    int ib   = wid << 4;                             // 16 rows per wave
    int m16  = lane & 15;
    float xi = inf[ib + m16];
    float ti = tgt[ib + m16];
    int sel  = (lane >> 4) << 1;                     // 0 for k=0/1 half, 2 for k=2/3 half

    v2f ones; ones.x = 1.0f; ones.y = 1.0f;          // B (4x16) = all ones
    v8f acc_a = 0.0f;
    v8f acc_b = 0.0f;

    for (int jb = 0; jb < NELEM; jb += TJ) {
        int j = jb + t;
        float wm = (lab[j] == 0) ? w[j] : 0.0f;
        xtw[t] = make_float4(inf[j], tgt[j], wm, 0.0f);
        __syncthreads();
        #pragma unroll 4
        for (int c = 0; c < TJ; c += 4) {
            float4 v0 = xtw[c + sel];       // broadcast within half-wave
            float4 v1 = xtw[c + sel + 1];
            v2f fa, fb;
            fa.x = fabsf(xi - v0.x) * v0.z; // |x_i - x_j| * wm_j
            fa.y = fabsf(xi - v1.x) * v1.z;
            fb.x = fabsf(ti - v0.y) * v0.z;
            fb.y = fabsf(ti - v1.y) * v1.z;
            acc_a = __builtin_amdgcn_wmma_f32_16x16x4_f32(false, fa, false, ones,
                                                          (short)0, acc_a, false, false);
            acc_b = __builtin_amdgcn_wmma_f32_16x16x4_f32(false, fb, false, ones,
                                                          (short)0, acc_b, false, false);
        }
        __syncthreads();
    }

    // D layout: VGPR r, lanes 0-15 -> m=r ; lanes 16-31 -> m=r+8 (replicated over n).
    if (m16 == 0) {
        int mo = (lane >> 4) * 8;
        float* arow = ws + WS_A;
        float* brow = ws + WS_B;
        #pragma unroll
        for (int r = 0; r < 8; ++r) {
            arow[ib + mo + r] = acc_a[r];
            brow[ib + mo + r] = acc_b[r];
        }
    }
}

// ---------------- K2: normalize rows in place, compute a_tot/b_tot ----------------
__global__ void k2_totals(const int* __restrict__ lab, const float* __restrict__ w,
                          float* __restrict__ ws) {
    __shared__ float sa[256];
    __shared__ float sb[256];
    int t = threadIdx.x;
    float inv = 1.0f / ws[WS_NF];
    float a_s = 0.0f, b_s = 0.0f;
    for (int i = t; i < NELEM; i += 256) {
        float wm = (lab[i] == 0) ? w[i] : 0.0f;
        float ar = ws[WS_A + i] * inv;   // \bar a_i = a_raw/nf
        float br = ws[WS_B + i] * inv;
        ws[WS_A + i] = ar;
        ws[WS_B + i] = br;
        a_s += ar * wm;
        b_s += br * wm;
    }
    sa[t] = a_s; sb[t] = b_s;
    __syncthreads();
    for (int o = 128; o > 0; o >>= 1) {
        if (t < o) { sa[t] += sa[t + o]; sb[t] += sb[t + o]; }
        __syncthreads();
    }
    if (t == 0) { ws[WS_ATOT] = sa[0] * inv; ws[WS_BTOT] = sb[0] * inv; }
}

// ---------------- K3: pass 2 — S_AB, S_AA, S_BB (nf^2 factors cancel) ----------------
__global__ void k3_pass2(const float* __restrict__ inf, const int* __restrict__ lab,
                         const float* __restrict__ tgt, const float* __restrict__ w,
                         float* __restrict__ ws) {
    __shared__ float4 tile[TJ];   // {x_j, t_j, wm_j, abar_j}
    __shared__ float  tbb[TJ];    // bbar_j
    __shared__ float  red[3 * 256];
    int t = threadIdx.x;
    int i = blockIdx.x * 256 + t;
    float xi  = inf[i];
    float ti  = tgt[i];
    float wmi = (lab[i] == 0) ? w[i] : 0.0f;
    float ca  = ws[WS_ATOT] - ws[WS_A + i];   // A_ij = a_ij + (ca - abar_j)
    float cb  = ws[WS_BTOT] - ws[WS_B + i];
    float s_ab = 0.0f, s_aa = 0.0f, s_bb = 0.0f;

    for (int jb = 0; jb < NELEM; jb += TJ) {
        int j = jb + t;
        float wm = (lab[j] == 0) ? w[j] : 0.0f;
        tile[t] = make_float4(inf[j], tgt[j], wm, ws[WS_A + j]);
        tbb[t]  = ws[WS_B + j];
        __syncthreads();
        #pragma unroll 4
        for (int jj = 0; jj < TJ; ++jj) {
            float4 v = tile[jj];
            float A  = fabsf(xi - v.x) + (ca - v.w);
            float B  = fabsf(ti - v.y) + (cb - tbb[jj]);
            float tA = v.z * A;
            s_ab = fmaf(tA, B, s_ab);
            s_aa = fmaf(tA, A, s_aa);
            s_bb = fmaf(v.z * B, B, s_bb);
        }
        __syncthreads();
    }
    red[t]       = s_ab * wmi;
    red[256 + t] = s_aa * wmi;
    red[512 + t] = s_bb * wmi;
    __syncthreads();
    for (int o = 128; o > 0; o >>= 1) {
        if (t < o) {
            red[t]       += red[t + o];
            red[256 + t] += red[256 + t + o];
            red[512 + t] += red[512 + t + o];
        }
        __syncthreads();
    }
    if (t == 0) {
        ws[WS_PART + blockIdx.x * 3 + 0] = red[0];
        ws[WS_PART + blockIdx.x * 3 + 1] = red[256];
        ws[WS_PART + blockIdx.x * 3 + 2] = red[512];
    }
}

// ---------------- K4: finalize ----------------
__global__ void k4_final(const float* __restrict__ ws, float* __restrict__ out) {
    if (threadIdx.x == 0) {
        float ab = 0.0f, aa = 0.0f, bb = 0.0f;
        for (int b = 0; b < NELEM / 256; ++b) {   // deterministic ordered sum
            ab += ws[WS_PART + b * 3 + 0];
            aa += ws[WS_PART + b * 3 + 1];
            bb += ws[WS_PART + b * 3 + 2];
        }
        float dcorr = (ab * ab) / (aa * bb);      // nf^2 factors cancelled
        float bce   = ws[WS_BCE];
        float ld    = 0.1f * dcorr;
        out[0] = bce + ld;
        out[1] = bce;
        out[2] = ld;
    }
}

extern "C" void kernel_launch(void* const* d_in, const int* in_sizes, int n_in,
                              void* d_out, int out_size, void* d_ws, size_t ws_size,
                              hipStream_t stream) {
    const float* inf = (const float*)d_in[0];
    const int*   lab = (const int*)d_in[1];
    const float* tgt = (const float*)d_in[2];
    const float* w   = (const float*)d_in[3];
    float* ws  = (float*)d_ws;
    float* out = (float*)d_out;

    k0_scalars<<<1, 256, 0, stream>>>(inf, lab, w, ws);
    k1_rowsums<<<NELEM / 128, 256, 0, stream>>>(inf, lab, tgt, w, ws);  // 16 rows/wave * 8 waves
    k2_totals<<<1, 256, 0, stream>>>(lab, w, ws);
    k3_pass2<<<NELEM / 256, 256, 0, stream>>>(inf, lab, tgt, w, ws);
    k4_final<<<1, 32, 0, stream>>>(ws, out);
}